// HybridQuanvolutionEstimator_65481071408063
// MI455X (gfx1250) — compile-verified
//
#include <hip/hip_runtime.h>

typedef __attribute__((ext_vector_type(2))) float v2f;
typedef __attribute__((ext_vector_type(8))) float v8f;

#define NPATCH 196
#define KDIM   784
#define TILE_M 16
#define KWAVES 4     // waves doing split-K GEMM
#define KSTEPS 49    // 196 k-steps of 4 / 4 waves

__global__ __launch_bounds__(256)
void quanv_fused_kernel(const float* __restrict__ x,       // [B,1,28,28]
                        const float* __restrict__ conv_w,  // [4,1,2,2]
                        const float* __restrict__ conv_b,  // [4]
                        const float* __restrict__ q_theta, // [4]
                        const float* __restrict__ lin_w,   // [10,784]
                        const float* __restrict__ lin_b,   // [10]
                        float* __restrict__ out)           // [B,10]
{
    __shared__ float qfTile[TILE_M * KDIM];   // 50176 B: 16 rows x 784 features
    __shared__ float ctile[KWAVES * 16 * 16]; // 4096 B: partial C tiles

    const int m0 = blockIdx.x * TILE_M;

    // Warm L2 with the (tiny, heavily reused) classifier weights.
    __builtin_prefetch(lin_w + ((int)threadIdx.x) * 32, 0, 1);

    // Uniform circuit parameters.
    const float th0 = 0.5f * q_theta[0], th1 = 0.5f * q_theta[1];
    const float th2 = 0.5f * q_theta[2], th3 = 0.5f * q_theta[3];
    const float g0c = cosf(th0), g0s = sinf(th0);
    const float g1c = cosf(th1), g1s = sinf(th1);
    const float g2c = cosf(th2), g2s = sinf(th2);
    const float g3c = cosf(th3), g3s = sinf(th3);

    float cw[16], cb[4];
    #pragma unroll
    for (int i = 0; i < 16; ++i) cw[i] = conv_w[i];
    #pragma unroll
    for (int i = 0; i < 4; ++i) cb[i] = conv_b[i];

    // ---------------- Phase 1: conv + 4-qubit statevector per patch -> LDS
    for (int t = threadIdx.x; t < TILE_M * NPATCH; t += blockDim.x) {
        const int row = t / NPATCH;
        const int p   = t - row * NPATCH;
        const float* xb = x + (size_t)(m0 + row) * 784;

        // 4 encoding angles for this patch (torch .view channel-scramble).
        float qc[4], qs[4];
        #pragma unroll
        for (int w = 0; w < 4; ++w) {
            const int idx = p * 4 + w;
            const int ch  = idx / 196;
            const int sp  = idx - ch * 196;
            const int r   = sp / 14;
            const int c   = sp - r * 14;
            const float* xw = xb + (2 * r) * 28 + 2 * c;
            const float a = cb[ch]
                          + xw[0]  * cw[ch * 4 + 0] + xw[1]  * cw[ch * 4 + 1]
                          + xw[28] * cw[ch * 4 + 2] + xw[29] * cw[ch * 4 + 3];
            const float h = 0.5f * a;
            qc[w] = cosf(h);
            qs[w] = sinf(h);
        }

        // Statevector: index i = q0*8 + q1*4 + q2*2 + q3.
        float re[16], im[16];
        #pragma unroll
        for (int i = 0; i < 16; ++i) {
            const float a0 = ((i >> 3) & 1) ? qs[0] : qc[0];
            const float a1 = ((i >> 2) & 1) ? qs[1] : qc[1];
            const float a2 = ((i >> 1) & 1) ? qs[2] : qc[2];
            const float a3 = (i & 1)        ? qs[3] : qc[3];
            re[i] = a0 * a1 * a2 * a3;
            im[i] = 0.0f;
        }
        // Ry(th0) on wire0 (mask 8): [[c,-s],[s,c]]
        #pragma unroll
        for (int i = 0; i < 16; ++i) if (!(i & 8)) {
            const int j = i | 8;
            const float r0 = g0c * re[i] - g0s * re[j];
            const float i0 = g0c * im[i] - g0s * im[j];
            const float r1 = g0s * re[i] + g0c * re[j];
            const float i1 = g0s * im[i] + g0c * im[j];
            re[i] = r0; im[i] = i0; re[j] = r1; im[j] = i1;
        }
        // Rx(th1) on wire1 (mask 4): [[c,-is],[-is,c]]
        #pragma unroll
        for (int i = 0; i < 16; ++i) if (!(i & 4)) {
            const int j = i | 4;
            const float r0 =  g1c * re[i] + g1s * im[j];
            const float i0 =  g1c * im[i] - g1s * re[j];
            const float r1 =  g1s * im[i] + g1c * re[j];
            const float i1 = -g1s * re[i] + g1c * im[j];
            re[i] = r0; im[i] = i0; re[j] = r1; im[j] = i1;
        }
        // Rz(th2) on wire2 (mask 2): diag(e^{-it}, e^{it})
        #pragma unroll
        for (int i = 0; i < 16; ++i) {
            const float r = re[i], q = im[i];
            if (i & 2) { re[i] = g2c * r - g2s * q; im[i] = g2c * q + g2s * r; }
            else       { re[i] = g2c * r + g2s * q; im[i] = g2c * q - g2s * r; }
        }
        // Ry(th3) on wire3 (mask 1)
        #pragma unroll
        for (int i = 0; i < 16; ++i) if (!(i & 1)) {
            const int j = i | 1;
            const float r0 = g3c * re[i] - g3s * re[j];
            const float i0 = g3c * im[i] - g3s * im[j];
            const float r1 = g3s * re[i] + g3c * re[j];
            const float i1 = g3s * im[i] + g3c * im[j];
            re[i] = r0; im[i] = i0; re[j] = r1; im[j] = i1;
        }
        // CNOT(0,1), CNOT(1,2), CNOT(2,3), CNOT(0,3): swap tgt within ctrl=1.
        #pragma unroll
        for (int i = 0; i < 16; ++i) if ((i & 8) && !(i & 4)) {
            const int j = i | 4;
            float tr = re[i]; re[i] = re[j]; re[j] = tr;
            float ti = im[i]; im[i] = im[j]; im[j] = ti;
        }
        #pragma unroll
        for (int i = 0; i < 16; ++i) if ((i & 4) && !(i & 2)) {
            const int j = i | 2;
            float tr = re[i]; re[i] = re[j]; re[j] = tr;
            float ti = im[i]; im[i] = im[j]; im[j] = ti;
        }
        #pragma unroll
        for (int i = 0; i < 16; ++i) if ((i & 2) && !(i & 1)) {
            const int j = i | 1;
            float tr = re[i]; re[i] = re[j]; re[j] = tr;
            float ti = im[i]; im[i] = im[j]; im[j] = ti;
        }
        #pragma unroll
        for (int i = 0; i < 16; ++i) if ((i & 8) && !(i & 1)) {
            const int j = i | 1;
            float tr = re[i]; re[i] = re[j]; re[j] = tr;
            float ti = im[i]; im[i] = im[j]; im[j] = ti;
        }
        // <Z_w> expectations.
        float e0 = 0.f, e1 = 0.f, e2 = 0.f, e3 = 0.f;
        #pragma unroll
        for (int i = 0; i < 16; ++i) {
            const float pr = re[i] * re[i] + im[i] * im[i];
            e0 += (i & 8) ? -pr : pr;
            e1 += (i & 4) ? -pr : pr;
            e2 += (i & 2) ? -pr : pr;
            e3 += (i & 1) ? -pr : pr;
        }
        *reinterpret_cast<float4*>(&qfTile[row * KDIM + p * 4]) =
            make_float4(e0, e1, e2, e3);
    }

    __syncthreads();

    // ---------------- Phase 2: split-K WMMA GEMM  C[16x16] = qfTile * lin_w^T
    const int wv   = threadIdx.x >> 5;
    const int lane = threadIdx.x & 31;

    if (wv < KWAVES) {                 // wave-uniform branch: EXEC all-ones inside
        const int l15   = lane & 15;         // A row (M) and B column (N)
        const int koff  = (lane >> 4) * 2;   // lanes 16-31 hold K+2/K+3
        const int kbase = wv * (KSTEPS * 4);
        const float* brow = lin_w + ((l15 < 10) ? l15 : 9) * KDIM;
        const float  bsel = (l15 < 10) ? 1.0f : 0.0f;

        v8f acc = {};
        for (int s = 0; s < KSTEPS; ++s) {
            const int k = kbase + s * 4 + koff;
            const float2 av = *reinterpret_cast<const float2*>(&qfTile[l15 * KDIM + k]);
            const float2 bv = *reinterpret_cast<const float2*>(brow + k);
            v2f a; a.x = av.x;        a.y = av.y;
            v2f b; b.x = bv.x * bsel; b.y = bv.y * bsel;
            acc = __builtin_amdgcn_wmma_f32_16x16x4_f32(
                false, a, false, b, (short)0, acc, false, false);
        }
        // Store partial tile: lanes 0-15 -> rows 0-7, lanes 16-31 -> rows 8-15.
        const int colN    = lane & 15;
        const int rowBase = (lane >> 4) * 8;
        #pragma unroll
        for (int r = 0; r < 8; ++r)
            ctile[wv * 256 + (rowBase + r) * 16 + colN] = acc[r];
    }

    __syncthreads();

    // ---------------- Phase 3: reduce split-K, bias, log_softmax, store.
    if (threadIdx.x < TILE_M) {
        const int row = threadIdx.x;
        float v[10];
        float mx = -3.402823466e38f;
        #pragma unroll
        for (int j = 0; j < 10; ++j) {
            float s = lin_b[j];
            #pragma unroll
            for (int w = 0; w < KWAVES; ++w) s += ctile[w * 256 + row * 16 + j];
            v[j] = s;
            mx = fmaxf(mx, s);
        }
        float sum = 0.f;
        #pragma unroll
        for (int j = 0; j < 10; ++j) sum += expf(v[j] - mx);
        const float lse = mx + logf(sum);
        float* o = out + (size_t)(m0 + row) * 10;
        #pragma unroll
        for (int j = 0; j < 10; ++j) o[j] = v[j] - lse;
    }
}

extern "C" void kernel_launch(void* const* d_in, const int* in_sizes, int n_in,
                              void* d_out, int out_size, void* d_ws, size_t ws_size,
                              hipStream_t stream) {
    const float* x       = (const float*)d_in[0];
    const float* conv_w  = (const float*)d_in[1];
    const float* conv_b  = (const float*)d_in[2];
    const float* q_theta = (const float*)d_in[3];
    const float* lin_w   = (const float*)d_in[4];
    const float* lin_b   = (const float*)d_in[5];
    float* out = (float*)d_out;

    const int B = in_sizes[0] / 784;   // x is [B,1,28,28]
    dim3 grid(B / TILE_M), block(256);
    quanv_fused_kernel<<<grid, block, 0, stream>>>(
        x, conv_w, conv_b, q_theta, lin_w, lin_b, out);

    (void)n_in; (void)out_size; (void)d_ws; (void)ws_size;
}